// TriplaneVAE_66434554134774
// MI455X (gfx1250) — compile-verified
//
#include <hip/hip_runtime.h>

// ---------------------------------------------------------------------------
// TriplaneVAE encoder for MI455X (gfx1250, wave32).
// Bandwidth-bound (~42 GFLOP vs ~1 GB traffic @ 23.3 TB/s HBM, plane
// accumulators L2-resident in 192 MB L2). fp32 WMMA (V_WMMA_F32_16X16X4_F32)
// for all GEMMs; pooling gather is FUSED into the GEMM kernel through LDS so
// the 205 MB concat buffer never touches HBM. Weights are staged in LDS
// pre-swizzled into per-lane fragment order so every B-fragment is one
// aligned ds_load_b64 (no VGPR repacking).
// ---------------------------------------------------------------------------

#define RESO   64
#define NSEG   (RESO * RESO)        // 4096
#define NBLK   5
#define BATCH  8
#define NPTS   100000
#define PTS    (BATCH * NPTS)       // 800000 = 6250 * 128

#define XT_STRIDE 68                // 128-row input tile, stride%64==4 -> conflict-free b64
#define LG_STRIDE 36                // 16-row transpose tile, stride%64==36 -> conflict-free b64

typedef __attribute__((ext_vector_type(2))) float v2f;
typedef __attribute__((ext_vector_type(8))) float v8f;

__device__ __forceinline__ v8f wmma4(v2f a, v2f b, v8f c) {
  // D = A(16x4) * B(4x16) + C, fp32.
  return __builtin_amdgcn_wmma_f32_16x16x4_f32(false, a, false, b, (short)0, c,
                                               false, false);
}

__device__ __forceinline__ v2f ld2(const float* p) {
  float2 t = *(const float2*)p;
  v2f r; r.x = t.x; r.y = t.y;
  return r;
}

// Stage a (K x 32) row-major weight matrix into LDS in per-lane fragment
// order: dst[(s*2 + tile)*32 + lane] = { w[kb][j], w[kb+1][j] } with
// kb = 4*s + 2*(lane>>4), j = tile*16 + (lane&15).  One ds_load_b64 per frag.
__device__ __forceinline__ void stage_swizzled(float2* dst, const float* w,
                                               int nfrag /*K/4*2*32*/, int tid) {
  for (int f = tid; f < nfrag; f += 256) {
    int lane = f & 31;
    int tile = (f >> 5) & 1;
    int s    = f >> 6;
    int kb   = 4 * s + 2 * (lane >> 4);
    int j    = tile * 16 + (lane & 15);
    float2 v;
    v.x = w[kb * 32 + j];
    v.y = w[(kb + 1) * 32 + j];
    dst[f] = v;
  }
}

// ---------------------------------------------------------------------------
// Kernel 1: plane indices per point
// ---------------------------------------------------------------------------
__global__ __launch_bounds__(256) void index_kernel(
    const float* __restrict__ pts, int* __restrict__ idx) {
  int p = blockIdx.x * 256 + threadIdx.x;
  if (p >= PTS) return;
  float px = pts[p * 3 + 0], py = pts[p * 3 + 1], pz = pts[p * 3 + 2];
  const float inv = 1.0f / (1.0f + 0.1f + 1e-3f);
  int xi = (int)fminf(fmaxf((px * inv + 1.0f) * (RESO * 0.5f), 0.0f), 63.0f);
  int yi = (int)fminf(fmaxf((py * inv + 1.0f) * (RESO * 0.5f), 0.0f), 63.0f);
  int zi = (int)fminf(fmaxf((pz * inv + 1.0f) * (RESO * 0.5f), 0.0f), 63.0f);
  idx[0 * PTS + p] = xi + RESO * yi;
  idx[1 * PTS + p] = xi + RESO * zi;
  idx[2 * PTS + p] = yi + RESO * zi;
}

// ---------------------------------------------------------------------------
// Kernel 2: fused (RFF-embed | pool-gather-concat) + ResnetBlockFC.
//   out = x @ sc + relu(relu(x) @ w0 + b0) @ w1 + b1
// MODE 0: x = [sin(2pi p@B), cos(2pi p@B)]  (block 0)
// MODE 1: x = [feat, sum_k gather(plane_mean_k)]  (blocks 1..4, in-place safe)
// Block = 8 waves = 128 points. Phase 1 builds x_tile in LDS, phase 2 does
// fp32 WMMA 16x16x4 on 16-row tiles per wave. The hidden activation g is
// transposed through the x_tile LDS region (reused after a barrier).
// ---------------------------------------------------------------------------
template <int MODE>
__global__ __launch_bounds__(256) void fused_block_kernel(
    const float* __restrict__ pts,      // MODE 0
    const float* __restrict__ rffB,     // MODE 0: (3,32)
    const float* __restrict__ feat_in,  // MODE 1: PTS x 32
    const int*   __restrict__ idx,      // MODE 1: 3 x PTS
    const float* __restrict__ sums,     // MODE 1: [3][B][NSEG][32]
    const float* __restrict__ icnt,     // MODE 1: [3][B][NSEG] (1/max(cnt,1))
    const float* __restrict__ w0,       // 64 x 32
    const float* __restrict__ b0,       // 32
    const float* __restrict__ w1,       // 32 x 32
    const float* __restrict__ b1,       // 32
    const float* __restrict__ scw,      // 64 x 32
    float* __restrict__ out)            // PTS x 32
{
  __shared__ float  xt[128 * XT_STRIDE];   // input tile / (reused) g transpose
  __shared__ float2 lw0s[1024];            // w0 fragments (16 steps x 2 tiles)
  __shared__ float2 lscs[1024];            // sc fragments
  __shared__ float2 lw1s[512];             // w1 fragments (8 steps x 2 tiles)
  __shared__ float  lpts[128 * 3];         // MODE 0 point cache
  __shared__ float  lrff[96];              // MODE 0 RFF matrix

  const int t    = threadIdx.x;
  const int wave = t >> 5;
  const int lane = t & 31;
  const int half = lane >> 4;
  const int mn   = lane & 15;
  const int blockbase = blockIdx.x * 128;

  stage_swizzled(lw0s, w0, 1024, t);
  stage_swizzled(lscs, scw, 1024, t);
  stage_swizzled(lw1s, w1, 512, t);

  if (MODE == 0) {
    for (int i = t; i < 128 * 3; i += 256) lpts[i] = pts[blockbase * 3 + i];
    if (t < 96) lrff[t] = rffB[t];
  }
  __syncthreads();

  // ---- phase 1: build x_tile[128][64] (channel j = lane, 8 points / iter)
  for (int it = 0; it < 16; ++it) {
    int pl = it * 8 + wave;        // local point 0..127
    int gp = blockbase + pl;       // global point
    if (MODE == 0) {
      float px = lpts[pl * 3 + 0], py = lpts[pl * 3 + 1], pz = lpts[pl * 3 + 2];
      const float TWO_PI = 6.283185307179586f;
      float pr = TWO_PI * (px * lrff[lane] + py * lrff[32 + lane] + pz * lrff[64 + lane]);
      float s, c;
      __sincosf(pr, &s, &c);
      xt[pl * XT_STRIDE + lane]      = s;
      xt[pl * XT_STRIDE + 32 + lane] = c;
    } else {
      int  b = gp / NPTS;
      float f = feat_in[gp * 32 + lane];
      float pooled = 0.0f;
#pragma unroll
      for (int k = 0; k < 3; ++k) {
        int seg = idx[k * PTS + gp];
        long base = ((long)k * BATCH + b) * NSEG + seg;
        pooled += sums[base * 32 + lane] * icnt[base];
      }
      xt[pl * XT_STRIDE + lane]      = f;
      xt[pl * XT_STRIDE + 32 + lane] = pooled;
    }
  }
  __syncthreads();

  // ---- phase 2: WMMA GEMMs (16 rows per wave)
  const int rowl = wave * 16;                 // local row base
  const float* arow = &xt[(rowl + mn) * XT_STRIDE];

  v8f accN0 = {}, accN1 = {};   // relu(x) @ w0
  v8f accS0 = {}, accS1 = {};   // x @ sc
#pragma unroll
  for (int s = 0; s < 16; ++s) {
    int kb = 4 * s + 2 * half;
    v2f a_raw = ld2(arow + kb);
    v2f a_rel; a_rel.x = fmaxf(a_raw.x, 0.0f); a_rel.y = fmaxf(a_raw.y, 0.0f);
    v2f b00 = ld2((const float*)&lw0s[(s * 2 + 0) * 32 + lane]);
    v2f b01 = ld2((const float*)&lw0s[(s * 2 + 1) * 32 + lane]);
    v2f bs0 = ld2((const float*)&lscs[(s * 2 + 0) * 32 + lane]);
    v2f bs1 = ld2((const float*)&lscs[(s * 2 + 1) * 32 + lane]);
    accN0 = wmma4(a_rel, b00, accN0);
    accN1 = wmma4(a_rel, b01, accN1);
    accS0 = wmma4(a_raw, bs0, accS0);
    accS1 = wmma4(a_raw, bs1, accS1);
  }

  // reuse x_tile region for the per-wave g transpose buffer
  __syncthreads();   // all waves done reading x_tile
  float* lg = &xt[wave * 16 * LG_STRIDE];   // [16][LG_STRIDE] per wave

  float bias0a = b0[mn], bias0b = b0[16 + mn];
#pragma unroll
  for (int v = 0; v < 8; ++v) {
    int r = v + half * 8;
    lg[r * LG_STRIDE + mn]      = fmaxf(accN0[v] + bias0a, 0.0f);
    lg[r * LG_STRIDE + 16 + mn] = fmaxf(accN1[v] + bias0b, 0.0f);
  }
  // same-wave LDS RAW: compiler inserts s_wait_dscnt; no cross-wave sharing.

  float bias1a = b1[mn], bias1b = b1[16 + mn];
  v8f acc0 = accS0, acc1 = accS1;
#pragma unroll
  for (int v = 0; v < 8; ++v) { acc0[v] += bias1a; acc1[v] += bias1b; }
#pragma unroll
  for (int s = 0; s < 8; ++s) {
    int kb = 4 * s + 2 * half;
    v2f a   = ld2(lg + mn * LG_STRIDE + kb);
    v2f bb0 = ld2((const float*)&lw1s[(s * 2 + 0) * 32 + lane]);
    v2f bb1 = ld2((const float*)&lw1s[(s * 2 + 1) * 32 + lane]);
    acc0 = wmma4(a, bb0, acc0);
    acc1 = wmma4(a, bb1, acc1);
  }

  float* orow = out + (long)(blockbase + rowl) * 32;
#pragma unroll
  for (int v = 0; v < 8; ++v) {
    int r = v + half * 8;
    orow[r * 32 + mn]      = acc0[v];
    orow[r * 32 + 16 + mn] = acc1[v];
  }
}

// ---------------------------------------------------------------------------
// Kernel 3: scatter-mean accumulation (atomics stay L2-resident: 12.6 MB)
// ---------------------------------------------------------------------------
__global__ __launch_bounds__(256) void scatter_kernel(
    const float* __restrict__ feat, const int* __restrict__ idx,
    float* __restrict__ sums, float* __restrict__ cnts) {
  long gid = (long)blockIdx.x * 256 + threadIdx.x;
  if (gid >= (long)PTS * 32) return;
  long p = gid >> 5;
  int  c = (int)(gid & 31);
  long b = p / NPTS;
  float v = feat[p * 32 + c];
#pragma unroll
  for (int k = 0; k < 3; ++k) {
    long base = ((long)k * BATCH + b) * NSEG + idx[(long)k * PTS + p];
    atomicAdd(&sums[base * 32 + c], v);
    if (c == 0) atomicAdd(&cnts[base], 1.0f);
  }
}

// ---------------------------------------------------------------------------
// Kernel 4: cnts -> 1/max(cnts,1) in place (98k rcp instead of 77M divides)
// ---------------------------------------------------------------------------
__global__ __launch_bounds__(256) void invcnt_kernel(float* __restrict__ cnts) {
  int i = blockIdx.x * 256 + threadIdx.x;
  if (i < 3 * BATCH * NSEG) cnts[i] = 1.0f / fmaxf(cnts[i], 1.0f);
}

// ---------------------------------------------------------------------------
// Kernel 5: final projection feat @ proj_w + proj_b  (WMMA, K = 32)
// ---------------------------------------------------------------------------
__global__ __launch_bounds__(256) void proj_kernel(
    const float* __restrict__ feat, const float* __restrict__ pw,
    const float* __restrict__ pb, float* __restrict__ out) {
  __shared__ float2 lws[512];
  int t = threadIdx.x;
  stage_swizzled(lws, pw, 512, t);
  __syncthreads();

  int wave = t >> 5, lane = t & 31, half = lane >> 4, mn = lane & 15;
  long rowbase = ((long)blockIdx.x * 8 + wave) * 16;

  v8f acc0 = {}, acc1 = {};
  float biasa = pb[mn], biasb = pb[16 + mn];
#pragma unroll
  for (int v = 0; v < 8; ++v) { acc0[v] = biasa; acc1[v] = biasb; }

  const float* xrow = feat + (rowbase + mn) * 32;
#pragma unroll
  for (int s = 0; s < 8; ++s) {
    int kb = 4 * s + 2 * half;
    v2f a   = ld2(xrow + kb);
    v2f b0v = ld2((const float*)&lws[(s * 2 + 0) * 32 + lane]);
    v2f b1v = ld2((const float*)&lws[(s * 2 + 1) * 32 + lane]);
    acc0 = wmma4(a, b0v, acc0);
    acc1 = wmma4(a, b1v, acc1);
  }

  float* orow = out + rowbase * 32;
#pragma unroll
  for (int v = 0; v < 8; ++v) {
    int r = v + half * 8;
    orow[r * 32 + mn]      = acc0[v];
    orow[r * 32 + 16 + mn] = acc1[v];
  }
}

// ---------------------------------------------------------------------------
// Kernel 6: planes: mean + transpose -> out[(k*B+b), c, seg]
// ---------------------------------------------------------------------------
__global__ __launch_bounds__(256) void finalize_kernel(
    const float* __restrict__ sums, const float* __restrict__ icnt,
    float* __restrict__ out) {
  long gid = (long)blockIdx.x * 256 + threadIdx.x;
  if (gid >= (long)3 * BATCH * 32 * NSEG) return;
  long seg  = gid & (NSEG - 1);
  long rest = gid >> 12;
  int  c    = (int)(rest & 31);
  long kb   = rest >> 5;                 // (k * BATCH + b)
  long base = kb * NSEG + seg;
  out[(kb * 32 + c) * NSEG + seg] = sums[base * 32 + c] * icnt[base];
}

// ---------------------------------------------------------------------------
extern "C" void kernel_launch(void* const* d_in, const int* in_sizes, int n_in,
                              void* d_out, int out_size, void* d_ws,
                              size_t ws_size, hipStream_t stream) {
  const float* input_pts = (const float*)d_in[0];  // (8, 100000, 3)
  const float* rff_B     = (const float*)d_in[1];  // (3, 32)
  const float* fc0_w     = (const float*)d_in[2];  // (5, 64, 32)
  const float* fc0_b     = (const float*)d_in[3];  // (5, 32)
  const float* fc1_w     = (const float*)d_in[4];  // (5, 32, 32)
  const float* fc1_b     = (const float*)d_in[5];  // (5, 32)
  const float* sc_w      = (const float*)d_in[6];  // (5, 64, 32)
  const float* proj_w    = (const float*)d_in[7];  // (32, 32)
  const float* proj_b    = (const float*)d_in[8];  // (32,)
  float* out = (float*)d_out;

  // workspace: idx (3*PTS int) | feat (PTS x 32) | sums | cnts   (~125 MB)
  char*  ws   = (char*)d_ws;
  int*   idx  = (int*)ws;
  float* feat = (float*)(ws + (size_t)3 * PTS * sizeof(int));
  float* sums = feat + (size_t)PTS * 32;
  float* cnts = sums + (size_t)3 * BATCH * NSEG * 32;
  size_t scatter_bytes =
      ((size_t)3 * BATCH * NSEG * 32 + (size_t)3 * BATCH * NSEG) * sizeof(float);

  float* out_planes = out;                                 // (24, 32, 64, 64)
  float* out_feat   = out + (size_t)3 * BATCH * 32 * NSEG; // (8, 100000, 32)

  dim3 blk(256);
  const int fused_blocks = PTS / 128;              // 6250 (exact)
  const int pc_blocks    = (int)(((long)PTS * 32 + 255) / 256);
  const int cnt_blocks   = (3 * BATCH * NSEG + 255) / 256;

  index_kernel<<<(PTS + 255) / 256, blk, 0, stream>>>(input_pts, idx);

  fused_block_kernel<0><<<fused_blocks, blk, 0, stream>>>(
      input_pts, rff_B, nullptr, nullptr, nullptr, nullptr,
      fc0_w, fc0_b, fc1_w, fc1_b, sc_w, feat);

  for (int i = 1; i < NBLK; ++i) {
    hipMemsetAsync(sums, 0, scatter_bytes, stream);
    scatter_kernel<<<pc_blocks, blk, 0, stream>>>(feat, idx, sums, cnts);
    invcnt_kernel<<<cnt_blocks, blk, 0, stream>>>(cnts);
    fused_block_kernel<1><<<fused_blocks, blk, 0, stream>>>(
        nullptr, nullptr, feat, idx, sums, cnts,
        fc0_w + (size_t)i * 64 * 32, fc0_b + (size_t)i * 32,
        fc1_w + (size_t)i * 32 * 32, fc1_b + (size_t)i * 32,
        sc_w + (size_t)i * 64 * 32, feat);
  }

  proj_kernel<<<PTS / 128, blk, 0, stream>>>(feat, proj_w, proj_b, out_feat);

  hipMemsetAsync(sums, 0, scatter_bytes, stream);
  scatter_kernel<<<pc_blocks, blk, 0, stream>>>(out_feat, idx, sums, cnts);
  invcnt_kernel<<<cnt_blocks, blk, 0, stream>>>(cnts);

  long fin_total = (long)3 * BATCH * 32 * NSEG;
  finalize_kernel<<<(int)((fin_total + 255) / 256), blk, 0, stream>>>(
      sums, cnts, out_planes);
}